// AbstractQCP_19258633355454
// MI455X (gfx1250) — compile-verified
//
#include <hip/hip_runtime.h>

#define Nn 4096
#define Mm 8192
#define KK 256

typedef float v2f __attribute__((ext_vector_type(2)));
typedef float v8f __attribute__((ext_vector_type(8)));

// ---- CDNA5 async global->LDS copy (ASYNCcnt-tracked), inline asm ----
__device__ __forceinline__ unsigned lds_off(const void* p) {
  // flat shared address: low 32 bits carry the LDS byte offset
  return (unsigned)(unsigned long long)p;
}
__device__ __forceinline__ void async_b128(unsigned loff, const void* g) {
  asm volatile("global_load_async_to_lds_b128 %0, %1, off"
               :: "v"(loff), "v"(g) : "memory");
}
__device__ __forceinline__ void wait_async_le12() {
  asm volatile("s_wait_asynccnt 0xc" ::: "memory");
}
__device__ __forceinline__ void wait_async_0() {
  asm volatile("s_wait_asynccnt 0x0" ::: "memory");
}

// A-frag (16x4 f32): lanes 0-15 -> M=lane, K={0,1}; lanes 16-31 -> K={2,3}.
// B-frag (4x16 f32): lanes 0-15 -> rows K={0,1}; lanes 16-31 -> K={2,3}; N=lane&15.
__device__ __forceinline__ v8f wmma4(v2f a, v2f b, v8f c) {
  return __builtin_amdgcn_wmma_f32_16x16x4_f32(false, a, false, b, (short)0, c,
                                               false, false);
}

#define KSTEP 64
#define PA_ITERS (Nn / KSTEP)          // 64
#define PB_ITERS (Mm / KSTEP)          // 128
#define TOT_ITERS (PA_ITERS + PB_ITERS)
#define A_STRIDE 68                    // 272 B row stride (16B-aligned)
#define B_STRIDE 132                   // 528 B row stride (16B-aligned)
#define A_WORDS (64 * A_STRIDE)        // [64 rows][64 data + pad]
#define B_WORDS (KSTEP * B_STRIDE)     // [64 rows][128 data + pad]

// out1[i,k] = sum_n P[i,n] wx[n,k] + sum_j A[j,i] mask[j] wy[j,k] + q[i] wt[k]
// Workgroup: 64 rows x 128 cols; 8 waves = 4 row tiles x 2 col halves.
__global__ __launch_bounds__(256) void qcp_out1_kernel(
    const float* __restrict__ P, const float* __restrict__ A,
    const float* __restrict__ q, const float* __restrict__ y,
    const float* __restrict__ s, const float* __restrict__ W,
    float* __restrict__ out) {
  __shared__ __align__(16) float Abuf[2][A_WORDS];
  __shared__ __align__(16) float Bbuf[2][B_WORDS];
  __shared__ float mskbuf[2][KSTEP];

  const int tid = threadIdx.x;
  const int lane = tid & 31;
  const int wv = tid >> 5;
  const int half = lane >> 4;
  const int l16 = lane & 15;
  const int rtb = (wv & 3) << 4;   // row-tile base within 64
  const int chb = (wv >> 2) << 6;  // col base within 128: 0 or 64
  const int i0 = blockIdx.x * 64;
  const int c0 = blockIdx.y * 128;

  const float* wx = W;
  const float* wy = W + (size_t)Nn * KK;
  const float* wt = W + (size_t)(Nn + Mm) * KK;

  v8f acc[4] = {};

  // Each call issues exactly 12 async b128 per wave (uniform ASYNCcnt).
  auto issue = [&](int it) {
    const int bsel = it & 1;
    float* Ab = &Abuf[bsel][0];
    float* Bb = &Bbuf[bsel][0];
    const bool phaseA = it < PA_ITERS;
    const int base = phaseA ? it * KSTEP : (it - PA_ITERS) * KSTEP;
    // A-operand slab: 64 rows x 64 floats, verbatim row copy.
    // phase A: rows of P (row i0+r, cols n0+c); phase B: rows of A ([j][i]).
    const float* Agsrc = phaseA ? (P + (size_t)(i0)*Nn + base)
                                : (A + (size_t)(base)*Nn + i0);
#pragma unroll
    for (int rep = 0; rep < 4; ++rep) {
      const int idx = rep * 256 + tid;
      const int r = idx >> 4, c = (idx & 15) << 2;
      async_b128(lds_off(Ab + r * A_STRIDE + c), Agsrc + (size_t)r * Nn + c);
    }
    const float* Bgsrc = phaseA ? (wx + (size_t)base * KK + c0)
                                : (wy + (size_t)base * KK + c0);
#pragma unroll
    for (int rep = 0; rep < 8; ++rep) {
      const int idx = rep * 256 + tid;
      const int r = idx >> 5, c = (idx & 31) << 2;
      async_b128(lds_off(Bb + r * B_STRIDE + c), Bgsrc + (size_t)r * KK + c);
    }
    if (!phaseA && tid < KSTEP) {
      const int jg = base + tid;
      mskbuf[bsel][tid] = (y[jg] - s[jg]) >= 0.f ? 1.f : 0.f;
    }
  };

  auto compute = [&](int it) {
    const int bsel = it & 1;
    const float* Ab = &Abuf[bsel][0];
    const float* Bb = &Bbuf[bsel][0];
    if (it < PA_ITERS) {
#pragma unroll
      for (int ks = 0; ks < KSTEP / 4; ++ks) {
        const int kb = (ks << 2) + (half << 1);
        v2f a;
        a.x = Ab[(rtb + l16) * A_STRIDE + kb + 0];
        a.y = Ab[(rtb + l16) * A_STRIDE + kb + 1];
#pragma unroll
        for (int ct = 0; ct < 4; ++ct) {
          const int cb = chb + (ct << 4) + l16;
          v2f b;
          b.x = Bb[(kb + 0) * B_STRIDE + cb];
          b.y = Bb[(kb + 1) * B_STRIDE + cb];
          acc[ct] = wmma4(a, b, acc[ct]);
        }
      }
    } else {
      const float* mk = &mskbuf[bsel][0];
#pragma unroll
      for (int ks = 0; ks < KSTEP / 4; ++ks) {
        const int kb = (ks << 2) + (half << 1);
        const float m0v = mk[kb], m1v = mk[kb + 1];
        v2f a;  // transposed read + mask folded into A-operand
        a.x = m0v * Ab[(kb + 0) * A_STRIDE + rtb + l16];
        a.y = m1v * Ab[(kb + 1) * A_STRIDE + rtb + l16];
#pragma unroll
        for (int ct = 0; ct < 4; ++ct) {
          const int cb = chb + (ct << 4) + l16;
          v2f b;
          b.x = Bb[(kb + 0) * B_STRIDE + cb];
          b.y = Bb[(kb + 1) * B_STRIDE + cb];
          acc[ct] = wmma4(a, b, acc[ct]);
        }
      }
    }
  };

  issue(0);
  for (int it = 0; it < TOT_ITERS; ++it) {
    const bool more = (it + 1) < TOT_ITERS;
    if (more) issue(it + 1);
    if (more) wait_async_le12(); else wait_async_0();
    __syncthreads();
    compute(it);
    __syncthreads();
  }

  // Epilogue: + q[i]*wt[k]
  const int rb = i0 + rtb + (half << 3);
  float qv[8];
#pragma unroll
  for (int r = 0; r < 8; ++r) qv[r] = q[rb + r];
#pragma unroll
  for (int ct = 0; ct < 4; ++ct) {
    const int col = c0 + chb + (ct << 4) + l16;
    const float wtv = wt[col];
#pragma unroll
    for (int r = 0; r < 8; ++r)
      out[(size_t)(rb + r) * KK + col] = acc[ct][r] + qv[r] * wtv;
  }
}

// out2[j,k] = b[j] wt[k] - sum_n A[j,n] wx[n,k] + (1-mask[j]) wy[j,k]
__global__ __launch_bounds__(256) void qcp_out2_kernel(
    const float* __restrict__ A, const float* __restrict__ b,
    const float* __restrict__ y, const float* __restrict__ s,
    const float* __restrict__ W, float* __restrict__ out2) {
  __shared__ __align__(16) float Abuf[2][A_WORDS];
  __shared__ __align__(16) float Bbuf[2][B_WORDS];

  const int tid = threadIdx.x;
  const int lane = tid & 31;
  const int wv = tid >> 5;
  const int half = lane >> 4;
  const int l16 = lane & 15;
  const int rtb = (wv & 3) << 4;
  const int chb = (wv >> 2) << 6;
  const int i0 = blockIdx.x * 64;  // rows within M
  const int c0 = blockIdx.y * 128;

  const float* wx = W;
  const float* wy = W + (size_t)Nn * KK;
  const float* wt = W + (size_t)(Nn + Mm) * KK;

  v8f acc[4] = {};

  auto issue = [&](int it) {
    const int bsel = it & 1;
    float* Ab = &Abuf[bsel][0];
    float* Bb = &Bbuf[bsel][0];
    const int n0 = it * KSTEP;
#pragma unroll
    for (int rep = 0; rep < 4; ++rep) {
      const int idx = rep * 256 + tid;
      const int r = idx >> 4, c = (idx & 15) << 2;
      async_b128(lds_off(Ab + r * A_STRIDE + c),
                 A + (size_t)(i0 + r) * Nn + n0 + c);
    }
#pragma unroll
    for (int rep = 0; rep < 8; ++rep) {
      const int idx = rep * 256 + tid;
      const int r = idx >> 5, c = (idx & 31) << 2;
      async_b128(lds_off(Bb + r * B_STRIDE + c),
                 wx + (size_t)(n0 + r) * KK + c0 + c);
    }
  };

  auto compute = [&](int it) {
    const int bsel = it & 1;
    const float* Ab = &Abuf[bsel][0];
    const float* Bb = &Bbuf[bsel][0];
#pragma unroll
    for (int ks = 0; ks < KSTEP / 4; ++ks) {
      const int kb = (ks << 2) + (half << 1);
      v2f a;
      a.x = Ab[(rtb + l16) * A_STRIDE + kb + 0];
      a.y = Ab[(rtb + l16) * A_STRIDE + kb + 1];
#pragma unroll
      for (int ct = 0; ct < 4; ++ct) {
        const int cb = chb + (ct << 4) + l16;
        v2f b;
        b.x = Bb[(kb + 0) * B_STRIDE + cb];
        b.y = Bb[(kb + 1) * B_STRIDE + cb];
        acc[ct] = wmma4(a, b, acc[ct]);
      }
    }
  };

  issue(0);
  for (int it = 0; it < PA_ITERS; ++it) {
    const bool more = (it + 1) < PA_ITERS;
    if (more) issue(it + 1);
    if (more) wait_async_le12(); else wait_async_0();
    __syncthreads();
    compute(it);
    __syncthreads();
  }

  const int rb = i0 + rtb + (half << 3);
  float bv[8], omk[8];
#pragma unroll
  for (int r = 0; r < 8; ++r) {
    bv[r] = b[rb + r];
    omk[r] = (y[rb + r] - s[rb + r]) >= 0.f ? 0.f : 1.f;
  }
#pragma unroll
  for (int ct = 0; ct < 4; ++ct) {
    const int col = c0 + chb + (ct << 4) + l16;
    const float wtv = wt[col];
#pragma unroll
    for (int r = 0; r < 8; ++r) {
      const int row = rb + r;
      out2[(size_t)row * KK + col] =
          bv[r] * wtv - acc[ct][r] + omk[r] * wy[(size_t)row * KK + col];
    }
  }
}

// Px = P @ x : one wave per row, lane-strided coalesced reads.
__global__ __launch_bounds__(256) void k_px(const float* __restrict__ P,
                                            const float* __restrict__ x,
                                            float* __restrict__ Px) {
  const int row = blockIdx.x * 8 + (threadIdx.x >> 5);
  const int lane = threadIdx.x & 31;
  float sum = 0.f;
  for (int n = lane; n < Nn; n += 32) sum += P[(size_t)row * Nn + n] * x[n];
#pragma unroll
  for (int off = 16; off > 0; off >>= 1) sum += __shfl_xor(sum, off, 32);
  if (lane == 0) Px[row] = sum;
}

__global__ __launch_bounds__(256) void k_xtpx(const float* __restrict__ x,
                                              const float* __restrict__ Px,
                                              float* __restrict__ outv) {
  __shared__ float red[256];
  const int tid = threadIdx.x;
  float sum = 0.f;
  for (int i = tid; i < Nn; i += 256) sum += x[i] * Px[i];
  red[tid] = sum;
  __syncthreads();
  for (int off = 128; off > 0; off >>= 1) {
    if (tid < off) red[tid] += red[tid + off];
    __syncthreads();
  }
  if (tid == 0) *outv = red[0];
}

__global__ __launch_bounds__(256) void k_rowred(
    const float* __restrict__ q, const float* __restrict__ b,
    const float* __restrict__ y, const float* __restrict__ s,
    const float* __restrict__ Px, const float* __restrict__ W,
    float* __restrict__ part) {
  const int t = threadIdx.x;
  const int row0 = blockIdx.x * 256;
  float sum = 0.f;
  for (int r = 0; r < 256; ++r) {
    const int rg = row0 + r;
    float coef;
    if (rg < Nn) {
      coef = -(q[rg] + 2.f * Px[rg]);
    } else {
      const int j = rg - Nn;
      const float mk = (y[j] - s[j]) >= 0.f ? 1.f : 0.f;
      coef = -b[j] * mk;
    }
    sum += coef * W[(size_t)rg * KK + t];
  }
  part[blockIdx.x * 256 + t] = sum;
}

__global__ __launch_bounds__(256) void k_out3(const float* __restrict__ W,
                                              const float* __restrict__ part,
                                              const float* __restrict__ xtpx,
                                              float* __restrict__ out3) {
  const int t = threadIdx.x;
  float sum = (*xtpx) * W[(size_t)(Nn + Mm) * KK + t];
  for (int bb = 0; bb < 48; ++bb) sum += part[bb * 256 + t];
  out3[t] = sum;
}

extern "C" void kernel_launch(void* const* d_in, const int* in_sizes, int n_in,
                              void* d_out, int out_size, void* d_ws,
                              size_t ws_size, hipStream_t stream) {
  (void)in_sizes; (void)n_in; (void)out_size; (void)ws_size;
  const float* P = (const float*)d_in[0];
  const float* A = (const float*)d_in[1];
  const float* q = (const float*)d_in[2];
  const float* b = (const float*)d_in[3];
  const float* x = (const float*)d_in[4];
  const float* y = (const float*)d_in[5];
  const float* s = (const float*)d_in[6];
  const float* W = (const float*)d_in[7];
  float* out = (float*)d_out;

  float* wsf = (float*)d_ws;
  float* Px = wsf;              // N floats
  float* xtpx = wsf + Nn;       // 1 float
  float* part = wsf + Nn + 64;  // 48*256 floats

  k_px<<<dim3(Nn / 8), 256, 0, stream>>>(P, x, Px);
  k_xtpx<<<1, 256, 0, stream>>>(x, Px, xtpx);
  qcp_out1_kernel<<<dim3(Nn / 64, 2), 256, 0, stream>>>(P, A, q, y, s, W, out);
  qcp_out2_kernel<<<dim3(Mm / 64, 2), 256, 0, stream>>>(
      A, b, y, s, W, out + (size_t)Nn * KK);
  k_rowred<<<48, 256, 0, stream>>>(q, b, y, s, Px, W, part);
  k_out3<<<1, 256, 0, stream>>>(W, part, xtpx, out + (size_t)(Nn + Mm) * KK);
}